// Myloss2_78323023610005
// MI455X (gfx1250) — compile-verified
//
#include <hip/hip_runtime.h>

// ---- types ----
typedef float f32x4 __attribute__((ext_vector_type(4)));
typedef int   i32x4 __attribute__((ext_vector_type(4)));
typedef float v2f   __attribute__((ext_vector_type(2)));
typedef float v8f   __attribute__((ext_vector_type(8)));

#define NBLOCKS 4096
#define NTHREADS 256

// per-element contribution: Loss + valid * (relu(lo-x)^2 + relu(x-hi)^2)
// SEG[i] = 1 - 0.25*i  =>  hi = 1 - 0.25*y, lo = hi - 0.25
__device__ __forceinline__ float elem_term(float xe, int ye, float le) {
    float yf   = (float)ye;
    float hi   = fmaf(-0.25f, yf, 1.0f);
    float lo   = hi - 0.25f;
    float amin = fmaxf(lo - xe, 0.0f);
    float amax = fmaxf(xe - hi, 0.0f);
    float h    = fmaf(amin, amin, amax * amax);
    return le + (((unsigned)ye < 4u) ? h : 0.0f);
}

// Kernel 1: streaming partial sums (HBM-bandwidth bound; NT B128 loads)
__global__ void loss_partial_kernel(const float* __restrict__ x,
                                    const int*   __restrict__ y,
                                    const float* __restrict__ loss,
                                    float* __restrict__ partial,
                                    long long n) {
    const long long n4     = n >> 2;
    const long long tid    = (long long)blockIdx.x * blockDim.x + threadIdx.x;
    const long long stride = (long long)gridDim.x * blockDim.x;

    const f32x4* x4 = (const f32x4*)x;
    const i32x4* y4 = (const i32x4*)y;
    const f32x4* l4 = (const f32x4*)loss;

    float s = 0.0f;
    for (long long i = tid; i < n4; i += stride) {
        f32x4 xv = __builtin_nontemporal_load(x4 + i);
        i32x4 yv = __builtin_nontemporal_load(y4 + i);
        f32x4 lv = __builtin_nontemporal_load(l4 + i);
        s += elem_term(xv.x, yv.x, lv.x);
        s += elem_term(xv.y, yv.y, lv.y);
        s += elem_term(xv.z, yv.z, lv.z);
        s += elem_term(xv.w, yv.w, lv.w);
    }
    // scalar tail (n not multiple of 4)
    for (long long i = (n4 << 2) + tid; i < n; i += stride)
        s += elem_term(x[i], y[i], loss[i]);

    // wave32 tree reduction (deterministic)
    for (int off = 16; off > 0; off >>= 1)
        s += __shfl_down(s, off, 32);

    __shared__ float smem[NTHREADS / 32];
    const int lane = threadIdx.x & 31;
    const int wid  = threadIdx.x >> 5;
    if (lane == 0) smem[wid] = s;
    __syncthreads();
    if (threadIdx.x == 0) {
        float b = 0.0f;
        const int nw = (blockDim.x + 31) >> 5;
        for (int w = 0; w < nw; ++w) b += smem[w];
        partial[blockIdx.x] = b;
    }
}

// Kernel 2: 256 threads (8 wave32s). Each thread issues 4 independent
// global_load_b128s (one loadcnt wait total, no serial latency chain),
// waves reduce by shuffle, and wave 0 (uniform branch -> EXEC all-ones,
// as WMMA requires) collapses the 8 wave sums with
// V_WMMA_F32_16X16X4_F32 using an all-ones B matrix (layout-invariant,
// exact fp32 FMA by 1.0):
//   D[m][n] = sum_k A[m][k] = v[lane m] + v[lane m+16]
// lane n's 8 D VGPRs hold rows {n<16 ? 0..7 : 8..15} of column n, so
// sum(d[0..7]) on lane0 + lane16 == total.
__global__ void final_reduce_wmma_kernel(const float* __restrict__ partial,
                                         float* __restrict__ out,
                                         long long n) {
    const int tid = threadIdx.x;              // 256 threads
    const f32x4* p4 = (const f32x4*)partial;  // NBLOCKS/4 = 1024 vec4s

    // 4 fixed-offset, independent vector loads per thread
    f32x4 v0 = p4[tid];
    f32x4 v1 = p4[tid + 256];
    f32x4 v2 = p4[tid + 512];
    f32x4 v3 = p4[tid + 768];
    f32x4 acc = (v0 + v1) + (v2 + v3);
    float s = (acc.x + acc.y) + (acc.z + acc.w);

    // wave32 tree reduction (deterministic)
    for (int off = 16; off > 0; off >>= 1)
        s += __shfl_down(s, off, 32);

    __shared__ float smem[8];
    const int lane = tid & 31;
    const int wid  = tid >> 5;
    if (lane == 0) smem[wid] = s;
    __syncthreads();

    if (wid == 0) {                 // uniform per wave: EXEC all-ones inside
        float v = smem[lane & 7];   // in-bounds for all lanes
        v = (lane < 8) ? v : 0.0f;  // v_cndmask, no EXEC change

        v2f a; a[0] = v;    a[1] = 0.0f;
        v2f b; b[0] = 1.0f; b[1] = 1.0f;
        v8f c = {};
        v8f d = __builtin_amdgcn_wmma_f32_16x16x4_f32(
            /*neg_a=*/false, a, /*neg_b=*/false, b,
            /*c_mod=*/(short)0, c, /*reuse_a=*/false, /*reuse_b=*/false);

        float t = 0.0f;
#pragma unroll
        for (int j = 0; j < 8; ++j) t += d[j];

        float t_hi = __shfl_down(t, 16, 32); // lane0 gets lane16's rows
        if (lane == 0) out[0] = (t + t_hi) / (float)n;
    }
}

extern "C" void kernel_launch(void* const* d_in, const int* in_sizes, int n_in,
                              void* d_out, int out_size, void* d_ws, size_t ws_size,
                              hipStream_t stream) {
    const float* x    = (const float*)d_in[0];
    const int*   y    = (const int*)d_in[1];
    const float* loss = (const float*)d_in[2];
    float* partial = (float*)d_ws;          // NBLOCKS floats = 16 KB
    float* out     = (float*)d_out;
    const long long n = (long long)in_sizes[0];

    loss_partial_kernel<<<NBLOCKS, NTHREADS, 0, stream>>>(x, y, loss, partial, n);
    final_reduce_wmma_kernel<<<1, NTHREADS, 0, stream>>>(partial, out, n);
}